// DepthSeparableConv2d_conv3_2_37572373905419
// MI455X (gfx1250) — compile-verified
//
#include <hip/hip_runtime.h>
#include <hip/hip_bf16.h>

typedef __attribute__((ext_vector_type(2))) float v2f;
typedef __attribute__((ext_vector_type(8))) float v8f;

#define BATCH 32
#define CIN   128
#define COUT  256
#define HW    3136      // 56*56
#define WDIM  56
#define NTILE 64        // spatial columns per pointwise block (3136 = 49*64)
#define NBLK  49

// ---------------------------------------------------------------- K0: zero the (b,o) max flags
__global__ void zero_flags_kernel(int* __restrict__ flags) {
    int i = blockIdx.x * 256 + threadIdx.x;
    if (i < BATCH * COUT) flags[i] = 0;   // 0 == bits of +0.0f
}

// ---------------------------------------------------------------- K1: depthwise 3x3 + bias + BN1 + ReLU + cut(4.0)
__global__ void __launch_bounds__(256)
dw_kernel(const float* __restrict__ x, const float* __restrict__ dw_w,
          const float* __restrict__ dw_b,
          const float* __restrict__ g1, const float* __restrict__ b1,
          const float* __restrict__ m1, const float* __restrict__ v1,
          float* __restrict__ y) {
    __shared__ float xs[HW];
    __shared__ float ys[HW];
    __shared__ float red[256];

    const int bc  = blockIdx.x;        // b*128 + c
    const int c   = bc & (CIN - 1);
    const int tid = threadIdx.x;

    const float* xp = x + (size_t)bc * HW;
    for (int i = tid; i < HW; i += 256) xs[i] = xp[i];

    const float w0 = dw_w[c * 9 + 0], w1 = dw_w[c * 9 + 1], w2 = dw_w[c * 9 + 2];
    const float w3 = dw_w[c * 9 + 3], w4 = dw_w[c * 9 + 4], w5 = dw_w[c * 9 + 5];
    const float w6 = dw_w[c * 9 + 6], w7 = dw_w[c * 9 + 7], w8 = dw_w[c * 9 + 8];
    const float scale = g1[c] * rsqrtf(v1[c] + 1e-5f);
    const float shift = b1[c] - m1[c] * scale;
    const float bias  = dw_b[c];

    __syncthreads();

    float lmax = 0.0f;
    for (int i = tid; i < HW; i += 256) {
        const int r  = i / WDIM;
        const int cc = i - r * WDIM;
        const int rm = r - 1, rp = r + 1, cm = cc - 1, cp = cc + 1;
        float acc = bias;
        if (rm >= 0) {
            const float* row = xs + rm * WDIM;
            if (cm >= 0)   acc += row[cm] * w0;
                           acc += row[cc] * w1;
            if (cp < WDIM) acc += row[cp] * w2;
        }
        {
            const float* row = xs + r * WDIM;
            if (cm >= 0)   acc += row[cm] * w3;
                           acc += row[cc] * w4;
            if (cp < WDIM) acc += row[cp] * w5;
        }
        if (rp < WDIM) {
            const float* row = xs + rp * WDIM;
            if (cm >= 0)   acc += row[cm] * w6;
                           acc += row[cc] * w7;
            if (cp < WDIM) acc += row[cp] * w8;
        }
        float val = fmaxf(acc * scale + shift, 0.0f);
        ys[i] = val;
        lmax = fmaxf(lmax, val);
    }

    red[tid] = lmax;
    __syncthreads();
    for (int s = 128; s > 0; s >>= 1) {
        if (tid < s) red[tid] = fmaxf(red[tid], red[tid + s]);
        __syncthreads();
    }
    const float keep = (red[0] >= 4.0f) ? 1.0f : 0.0f;

    float* yp = y + (size_t)bc * HW;
    for (int i = tid; i < HW; i += 256) yp[i] = ys[i] * keep;
}

// ---------------------------------------------------------------- K2: pointwise 1x1 GEMM via V_WMMA_F32_16X16X4_F32
// Per batch b:  Z[256 x 3136] = W[256 x 128] * Y[128 x 3136], fused bias+BN2+ReLU,
// plus atomicMax per (b,o) for the 0.001 cut.
__global__ void __launch_bounds__(256)
pw_wmma_kernel(const float* __restrict__ y, const float* __restrict__ pw_w,
               const float* __restrict__ pw_b,
               const float* __restrict__ g2, const float* __restrict__ b2,
               const float* __restrict__ m2, const float* __restrict__ v2,
               float* __restrict__ z, int* __restrict__ zmax) {
    // LDS y-tile, swizzled so B fragments are contiguous v2f: addr(k,n) = (k>>2)*256 + n*4 + (k&3)
    __shared__ float ylds[CIN * NTILE];

    const int blk = blockIdx.x;
    const int b   = blk / NBLK;
    const int hw0 = (blk - b * NBLK) * NTILE;
    const int tid = threadIdx.x;

    const float* yb = y + (size_t)b * CIN * HW + hw0;
    for (int e = tid; e < CIN * NTILE; e += 256) {
        const int c = e >> 6;           // 0..127
        const int n = e & 63;           // 0..63
        ylds[((c >> 2) << 8) + (n << 2) + (c & 3)] = yb[(size_t)c * HW + n];
    }
    __syncthreads();

    const int lane  = tid & 31;
    const int wv    = tid >> 5;         // 8 waves
    const int mbase = wv * 32;          // this wave: output channels [mbase, mbase+32)
    const int l15   = lane & 15;
    const int hi    = lane >> 4;        // 0 / 1 (upper lane half -> K+2 per ISA A/B layout)

    v8f acc[2][4] = {};                 // 2 M-tiles x 4 N-tiles of 16x16 f32

    // A(m,k) = pw_w[m*128 + k]; lane holds row mbase+t*16+l15, v2f covers K = k0+2*hi+{0,1}
    const float* wrow0 = pw_w + (size_t)(mbase + l15) * CIN + 2 * hi;
    const float* wrow1 = wrow0 + 16 * CIN;

    for (int k0 = 0; k0 < CIN; k0 += 4) {
        const v2f a0 = *(const v2f*)(wrow0 + k0);
        const v2f a1 = *(const v2f*)(wrow1 + k0);
        v2f bf[4];
#pragma unroll
        for (int j = 0; j < 4; ++j)
            bf[j] = *(const v2f*)(&ylds[((k0 >> 2) << 8) + ((j * 16 + l15) << 2) + 2 * hi]);
#pragma unroll
        for (int j = 0; j < 4; ++j) {
            acc[0][j] = __builtin_amdgcn_wmma_f32_16x16x4_f32(
                false, a0, false, bf[j], (short)0, acc[0][j], false, false);
            acc[1][j] = __builtin_amdgcn_wmma_f32_16x16x4_f32(
                false, a1, false, bf[j], (short)0, acc[1][j], false, false);
        }
    }

    // Epilogue: C/D layout -> vgpr r, lanes0-15: (M=r, N=lane); lanes16-31: (M=r+8, N=lane-16)
    float* zb = z + (size_t)b * COUT * HW;
#pragma unroll
    for (int t = 0; t < 2; ++t) {
#pragma unroll
        for (int r = 0; r < 8; ++r) {
            const int o = mbase + t * 16 + r + hi * 8;
            const float scale = g2[o] * rsqrtf(v2[o] + 1e-5f);
            const float shift = b2[o] - m2[o] * scale;
            const float bias  = pw_b[o];
            float omax = 0.0f;
#pragma unroll
            for (int j = 0; j < 4; ++j) {
                float val = acc[t][j][r] + bias;
                val = fmaxf(val * scale + shift, 0.0f);
                zb[(size_t)o * HW + hw0 + j * 16 + l15] = val;
                omax = fmaxf(omax, val);
            }
            // max across the 16 lanes sharing this o (stay inside the half-wave)
            for (int sh = 1; sh < 16; sh <<= 1)
                omax = fmaxf(omax, __shfl_xor(omax, sh, 32));
            if (l15 == 0)
                atomicMax(&zmax[b * COUT + o], __float_as_int(omax)); // vals >= 0 -> int order == float order
        }
    }
}

// ---------------------------------------------------------------- K3: apply cut(0.001) on z
__global__ void __launch_bounds__(256)
cut_kernel(float* __restrict__ z, const int* __restrict__ zmax) {
    const int i4 = blockIdx.x * 256 + threadIdx.x;       // one float4 each; HW % 4 == 0
    const int e  = i4 * 4;
    const int bo = e / HW;                               // b*256 + o (same for all 4 elems)
    const float mx = __int_as_float(zmax[bo]);
    if (mx < 0.001f) {
        float4 zero = {0.f, 0.f, 0.f, 0.f};
        ((float4*)z)[i4] = zero;
    }
}

// ---------------------------------------------------------------- launch
extern "C" void kernel_launch(void* const* d_in, const int* in_sizes, int n_in,
                              void* d_out, int out_size, void* d_ws, size_t ws_size,
                              hipStream_t stream) {
    (void)in_sizes; (void)n_in; (void)out_size; (void)ws_size;

    const float* x    = (const float*)d_in[0];
    const float* dw_w = (const float*)d_in[1];
    const float* dw_b = (const float*)d_in[2];
    const float* g1   = (const float*)d_in[3];
    const float* b1   = (const float*)d_in[4];
    const float* m1   = (const float*)d_in[5];
    const float* v1   = (const float*)d_in[6];
    const float* pw_w = (const float*)d_in[7];
    const float* pw_b = (const float*)d_in[8];
    const float* g2   = (const float*)d_in[9];
    const float* b2   = (const float*)d_in[10];
    const float* m2   = (const float*)d_in[11];
    const float* v2   = (const float*)d_in[12];
    float* z = (float*)d_out;

    // workspace layout: y [32*128*3136 f32] then zmax flags [32*256 i32]
    float* y_ws  = (float*)d_ws;
    int*   zmax  = (int*)((char*)d_ws + (size_t)BATCH * CIN * HW * sizeof(float));

    zero_flags_kernel<<<(BATCH * COUT + 255) / 256, 256, 0, stream>>>(zmax);

    dw_kernel<<<BATCH * CIN, 256, 0, stream>>>(x, dw_w, dw_b, g1, b1, m1, v1, y_ws);

    pw_wmma_kernel<<<BATCH * NBLK, 256, 0, stream>>>(y_ws, pw_w, pw_b,
                                                     g2, b2, m2, v2, z, zmax);

    const int total4 = (BATCH * COUT * HW) / 4;          // 6,422,528
    cut_kernel<<<total4 / 256, 256, 0, stream>>>(z, zmax);
}